// DownSample_38276748542410
// MI455X (gfx1250) — compile-verified
//
#include <hip/hip_runtime.h>
#include <math.h>

typedef __attribute__((ext_vector_type(2))) float v2f;
typedef __attribute__((ext_vector_type(8))) float v8f;

#define BN_SCALE 0.9999950000374997f   // 1/sqrt(1 + 1e-5)

__device__ __forceinline__ float gelu_exact(float x) {
    return 0.5f * x * (1.0f + erff(x * 0.7071067811865475f));
}

__device__ __forceinline__ v8f wmma_f32(v2f a, v2f b, v8f c) {
    // D = A(16x4) * B(4x16) + C, all f32, wave32
    return __builtin_amdgcn_wmma_f32_16x16x4_f32(false, a, false, b, (short)0, c, false, false);
}

// ---------------------------------------------------------------------------
// Kernel 1: FPS (m=16) + 2-NN + coor_out gather.  1 wave per batch, lane=point.
// ---------------------------------------------------------------------------
__global__ __launch_bounds__(32) void k_fps_knn(
    const float* __restrict__ stk,    // [128][32][32]
    int* __restrict__ fps_ws,         // [128][16]
    int* __restrict__ knn_ws,         // [128][16][2]
    float* __restrict__ coor_out)     // [128][16][32]
{
    const int b = blockIdx.x;
    const int lane = threadIdx.x;     // point index 0..31
    __shared__ float pts[32 * 33];
    __shared__ float sqs[32];
    __shared__ int   fps_s[16];
    __shared__ int   krow[64];

    float x[32];
    const float* base = stk + (b * 32 + lane) * 32;
    float sq = 0.f;
#pragma unroll
    for (int c = 0; c < 32; ++c) {
        x[c] = base[c];
        pts[lane * 33 + c] = x[c];
        sq += x[c] * x[c];
    }
    sqs[lane] = sq;
    __syncthreads();

    // ---- farthest point sampling (deterministic start at 0) ----
    float dist = INFINITY;
    int far = 0;
#pragma unroll 1
    for (int it = 0; it < 16; ++it) {
        if (lane == 0) fps_s[it] = far;
        float d = 0.f;
#pragma unroll
        for (int c = 0; c < 32; ++c) {
            float t = x[c] - pts[far * 33 + c];
            d += t * t;
        }
        dist = fminf(dist, d);
        // wave32 argmax, first (lowest-index) max wins -> matches jnp.argmax
        float bv = dist; int bi = lane;
#pragma unroll
        for (int off = 16; off; off >>= 1) {
            float ov = __shfl_xor(bv, off, 32);
            int   oi = __shfl_xor(bi, off, 32);
            if (ov > bv || (ov == bv && oi < bi)) { bv = ov; bi = oi; }
        }
        far = bi;
    }
    __syncthreads();

    // ---- 2-NN per row (incl. self); stable tie-break = lower index ----
    float b0v = INFINITY, b1v = INFINITY;
    int b0i = 0, b1i = 0;
#pragma unroll 1
    for (int j = 0; j < 32; ++j) {
        float dot = 0.f;
#pragma unroll
        for (int c = 0; c < 32; ++c) dot += x[c] * pts[j * 33 + c];
        float d = sq + sqs[j] - 2.f * dot;
        if (d < b0v)      { b1v = b0v; b1i = b0i; b0v = d; b0i = j; }
        else if (d < b1v) { b1v = d; b1i = j; }
    }
    krow[lane * 2 + 0] = b0i;
    krow[lane * 2 + 1] = b1i;
    __syncthreads();

    if (lane < 16) {
        int s0 = fps_s[lane];
        fps_ws[b * 16 + lane] = s0;
        knn_ws[(b * 16 + lane) * 2 + 0] = krow[s0 * 2 + 0];
        knn_ws[(b * 16 + lane) * 2 + 1] = krow[s0 * 2 + 1];
    }
#pragma unroll 1
    for (int mm = 0; mm < 16; ++mm) {
        coor_out[(b * 16 + mm) * 32 + lane] = pts[fps_s[mm] * 33 + lane];
    }
}

// ---------------------------------------------------------------------------
// Kernel 2: sparse branch.  One block per batch; GEMM [256x512]x[512x32].
// Columns col = n*16 + m so the max over n is an in-register tile max.
// ---------------------------------------------------------------------------
__global__ __launch_bounds__(256) void k_sparse(
    const float* __restrict__ sfea,   // [128][256][32]
    const float* __restrict__ sp_w,   // [256][512]
    const float* __restrict__ sp_b,
    const float* __restrict__ sp_g,
    const float* __restrict__ sp_be,
    const int*  __restrict__ fps_ws,
    const int*  __restrict__ knn_ws,
    float* __restrict__ sparse_out)   // [128][256][16]
{
    const int b = blockIdx.x;
    const int tid = threadIdx.x;
    __shared__ __align__(16) float XT[32 * 516];   // [col][q], padded

    const float* sb = sfea + b * 256 * 32;
#pragma unroll 1
    for (int e = tid; e < 32 * 512; e += 256) {
        int q = e >> 5;
        int col = e & 31;
        int n = col >> 4, m = col & 15;
        int s0 = fps_ws[b * 16 + m];
        float v;
        if (q < 256) {
            int sn = knn_ws[(b * 16 + m) * 2 + n];
            v = sb[q * 32 + sn] - sb[q * 32 + s0];       // sa = sn - sc
        } else {
            v = sb[(q - 256) * 32 + s0];                  // sc broadcast
        }
        XT[col * 516 + q] = v;
    }
    __syncthreads();

    const int wv = tid >> 5, lane = tid & 31;
    const int fr = lane & 15, fh = lane >> 4;
    v8f acc00 = {}, acc01 = {}, acc10 = {}, acc11 = {};
    const float* a0p = sp_w + (wv * 16 + fr) * 512;
    const float* a1p = sp_w + ((wv + 8) * 16 + fr) * 512;
#pragma unroll 4
    for (int k = 0; k < 512; k += 4) {
        int kq = k + 2 * fh;
        v2f a0 = *(const v2f*)(a0p + kq);
        v2f a1 = *(const v2f*)(a1p + kq);
        v2f b0 = *(const v2f*)(&XT[fr * 516 + kq]);
        v2f b1 = *(const v2f*)(&XT[(16 + fr) * 516 + kq]);
        acc00 = wmma_f32(a0, b0, acc00);
        acc01 = wmma_f32(a0, b1, acc01);
        acc10 = wmma_f32(a1, b0, acc10);
        acc11 = wmma_f32(a1, b1, acc11);
    }
#pragma unroll
    for (int mt = 0; mt < 2; ++mt) {
        v8f p0 = mt ? acc10 : acc00;
        v8f p1 = mt ? acc11 : acc01;
        int Mtile = wv + 8 * mt;
#pragma unroll
        for (int v = 0; v < 8; ++v) {
            int o = Mtile * 16 + v + 8 * fh;
            float bias = sp_b[o], g = sp_g[o], be = sp_be[o];
            float y0 = gelu_exact(g * ((p0[v] + bias) * BN_SCALE) + be);
            float y1 = gelu_exact(g * ((p1[v] + bias) * BN_SCALE) + be);
            sparse_out[(b * 256 + o) * 16 + fr] = fmaxf(y0, y1);
        }
    }
}

// ---------------------------------------------------------------------------
// Kernel 3: dense branch.  One block per (b,m).
//   GEMM1: [128x256] x [256x64]  (col = n*32 + p -> in-register max over n)
//   conv as GEMM2: A = ds_w [o][i*3+t] (contiguous), B from zero-padded d4 LDS
// ---------------------------------------------------------------------------
__global__ __launch_bounds__(256) void k_dense(
    const float* __restrict__ dfea,   // [128][128][32][32]
    const float* __restrict__ dn_w,   // [128][256]
    const float* __restrict__ dn_b,
    const float* __restrict__ dn_g,
    const float* __restrict__ dn_be,
    const float* __restrict__ ds_w,   // [128][128][1][3] -> [o][384]
    const float* __restrict__ ds_b,
    const float* __restrict__ ds_g,
    const float* __restrict__ ds_be,
    const int*  __restrict__ fps_ws,
    const int*  __restrict__ knn_ws,
    float* __restrict__ dense_out)    // [128][128][16][16]
{
    const int blk = blockIdx.x;
    const int b = blk >> 4, m = blk & 15;
    const int tid = threadIdx.x;
    __shared__ __align__(16) float XT[64 * 260];   // [col][q]
    __shared__ float D4[128 * 36];                 // [o][sp+1], col 0 = zero pad

    const int s0 = fps_ws[b * 16 + m];
    const int s1 = knn_ws[(b * 16 + m) * 2 + 0];
    const int s2 = knn_ws[(b * 16 + m) * 2 + 1];
    const float* db = dfea + b * 128 * 32 * 32;    // [c][s][sp]

    if (tid < 128) D4[tid * 36] = 0.f;             // conv left zero pad

    // Build X[q, col]: exact torch-reshape semantics of the reference.
#pragma unroll 1
    for (int e = tid; e < 256 * 64; e += 256) {
        int q = e >> 6;
        int col = e & 63;
        int n = col >> 5, p = col & 31;
        int c = q & 127, hi = q >> 7;
        int sn = n ? s2 : s1;
        float v;
        if (p < 16) {
            int sp = 2 * p + hi;
            v = db[(c * 32 + sn) * 32 + sp] - db[(c * 32 + s0) * 32 + sp];
        } else {
            int sp = 2 * (p - 16) + hi;
            v = db[(c * 32 + s0) * 32 + sp];
        }
        XT[col * 260 + q] = v;
    }
    __syncthreads();

    const int wv = tid >> 5, lane = tid & 31;
    const int fr = lane & 15, fh = lane >> 4;
    v8f acc0 = {}, acc1 = {}, acc2 = {}, acc3 = {};
    const float* ap = dn_w + (wv * 16 + fr) * 256;
#pragma unroll 4
    for (int k = 0; k < 256; k += 4) {
        int kq = k + 2 * fh;
        v2f a  = *(const v2f*)(ap + kq);
        v2f b0 = *(const v2f*)(&XT[fr * 260 + kq]);
        v2f b1 = *(const v2f*)(&XT[(16 + fr) * 260 + kq]);
        v2f b2 = *(const v2f*)(&XT[(32 + fr) * 260 + kq]);
        v2f b3 = *(const v2f*)(&XT[(48 + fr) * 260 + kq]);
        acc0 = wmma_f32(a, b0, acc0);
        acc1 = wmma_f32(a, b1, acc1);
        acc2 = wmma_f32(a, b2, acc2);
        acc3 = wmma_f32(a, b3, acc3);
    }
    // bias + BN + exact GELU, max over n (tiles {0,1} vs {2,3}), d4 -> LDS
#pragma unroll
    for (int v = 0; v < 8; ++v) {
        int o = wv * 16 + v + 8 * fh;
        float bias = dn_b[o], g = dn_g[o], be = dn_be[o];
        float y00 = gelu_exact(g * ((acc0[v] + bias) * BN_SCALE) + be);
        float y01 = gelu_exact(g * ((acc2[v] + bias) * BN_SCALE) + be);
        float y10 = gelu_exact(g * ((acc1[v] + bias) * BN_SCALE) + be);
        float y11 = gelu_exact(g * ((acc3[v] + bias) * BN_SCALE) + be);
        D4[o * 36 + 1 + fr]      = fmaxf(y00, y01);   // p = fr
        D4[o * 36 + 1 + 16 + fr] = fmaxf(y10, y11);   // p = 16 + fr
    }
    __syncthreads();

    // conv 1x3 stride 2 as WMMA GEMM: K = 384 = (i,t); sp = 2w-1+t -> D4 col 2w+t
    v8f cacc = {};
    const float* cap = ds_w + (wv * 16 + fr) * 384;
#pragma unroll 2
    for (int k = 0; k < 384; k += 4) {
        int kk = k + 2 * fh;
        v2f a = *(const v2f*)(cap + kk);
        int i0 = kk / 3,       t0 = kk - 3 * i0;
        int i1 = (kk + 1) / 3, t1 = (kk + 1) - 3 * i1;
        v2f bf;
        bf.x = D4[i0 * 36 + 2 * fr + t0];
        bf.y = D4[i1 * 36 + 2 * fr + t1];
        cacc = wmma_f32(a, bf, cacc);
    }
#pragma unroll
    for (int v = 0; v < 8; ++v) {
        int o = wv * 16 + v + 8 * fh;
        float pre = cacc[v] + ds_b[o];
        float y = gelu_exact(ds_g[o] * (pre * BN_SCALE) + ds_be[o]);
        dense_out[((b * 128 + o) * 16 + m) * 16 + fr] = y;
    }
}

// ---------------------------------------------------------------------------
extern "C" void kernel_launch(void* const* d_in, const int* in_sizes, int n_in,
                              void* d_out, int out_size, void* d_ws, size_t ws_size,
                              hipStream_t stream) {
    const float* sparse_fea = (const float*)d_in[0];
    const float* dense_fea  = (const float*)d_in[1];
    const float* stk_coor   = (const float*)d_in[2];
    // d_in[3] = n_stk_center (16), baked into launch geometry
    const float* sp_w  = (const float*)d_in[4];
    const float* sp_b  = (const float*)d_in[5];
    const float* sp_g  = (const float*)d_in[6];
    const float* sp_be = (const float*)d_in[7];
    const float* dn_w  = (const float*)d_in[8];
    const float* dn_b  = (const float*)d_in[9];
    const float* dn_g  = (const float*)d_in[10];
    const float* dn_be = (const float*)d_in[11];
    const float* ds_w  = (const float*)d_in[12];
    const float* ds_b  = (const float*)d_in[13];
    const float* ds_g  = (const float*)d_in[14];
    const float* ds_be = (const float*)d_in[15];

    float* out        = (float*)d_out;
    float* sparse_out = out;                                   // 128*256*16
    float* dense_out  = out + 128 * 256 * 16;                  // 128*128*16*16
    float* coor_out   = out + 128 * 256 * 16 + 128 * 128 * 16 * 16;  // 128*16*32

    int* fps_ws = (int*)d_ws;            // [128][16]
    int* knn_ws = fps_ws + 128 * 16;     // [128][16][2]

    k_fps_knn<<<128, 32, 0, stream>>>(stk_coor, fps_ws, knn_ws, coor_out);
    k_sparse<<<128, 256, 0, stream>>>(sparse_fea, sp_w, sp_b, sp_g, sp_be,
                                      fps_ws, knn_ws, sparse_out);
    k_dense<<<2048, 256, 0, stream>>>(dense_fea, dn_w, dn_b, dn_g, dn_be,
                                      ds_w, ds_b, ds_g, ds_be,
                                      fps_ws, knn_ws, dense_out);
}